// Generator_38474317038072
// MI455X (gfx1250) — compile-verified
//
#include <hip/hip_runtime.h>
#include <hip/hip_bf16.h>

// ---------------------------------------------------------------------------
// Problem constants (from reference)
// ---------------------------------------------------------------------------
#define BB   16384
#define TT   100
#define ZDIM 128
#define HH   4096
#define NEG_SLOPE 0.01f

typedef __attribute__((ext_vector_type(2))) float v2f;
typedef __attribute__((ext_vector_type(8))) float v8f;
typedef __attribute__((ext_vector_type(4))) int   i4;   // b128 payload type

__device__ __forceinline__ float lrelu(float v) {
    return v >= 0.0f ? v : NEG_SLOPE * v;
}

// ---------------------------------------------------------------------------
// CDNA5 async direct-to-LDS staging (GLOBAL_LOAD_ASYNC_TO_LDS_B128, ASYNCcnt).
// Builtin signature (from hipcc diagnostic): takes int4 addrspace(1)* source
// and int4 addrspace(3)* LDS destination plus imm offset / imm cpol.
// Guarded so the file still compiles (reg-staged fallback) on toolchains
// without the gfx1250 async builtins.
// ---------------------------------------------------------------------------
#if __has_builtin(__builtin_amdgcn_global_load_async_to_lds_b128) && \
    __has_builtin(__builtin_amdgcn_s_wait_asynccnt)
#define USE_ASYNC_LDS 1
#else
#define USE_ASYNC_LDS 0
#endif

__device__ __forceinline__ void copy16_to_lds(const float* __restrict__ src,
                                              float* __restrict__ dst_lds) {
#if USE_ASYNC_LDS
    __builtin_amdgcn_global_load_async_to_lds_b128(
        (__attribute__((address_space(1))) i4*)(const_cast<float*>(src)),
        (__attribute__((address_space(3))) i4*)(dst_lds),
        /*offset=*/0, /*cpol=*/0);
#else
    *(float4*)dst_lds = *(const float4*)src;
#endif
}

template <int N>
__device__ __forceinline__ void wait_async() {
#if USE_ASYNC_LDS
    __builtin_amdgcn_s_wait_asynccnt((unsigned short)N);  // s_wait_asynccnt N
#endif
}

// ---------------------------------------------------------------------------
// Fused GEMM + bias + LeakyReLU using V_WMMA_F32_16X16X4_F32 (wave32).
//   C[M,N] = lrelu(A[M,K] * B[K,N] + bias[N])
// Block tile 128x128, K-tile 32, DOUBLE-BUFFERED LDS fed by async-to-LDS
// loads: tile i+1's 8 B128 loads per wave are issued, then
// s_wait_asynccnt 8 retires tile i's loads while i+1 stays in flight.
// 256 threads = 8 waves in a 4(M) x 2(N) grid; each wave owns a 32x64
// subtile = 2x4 WMMA 16x16 accumulators.
// M % 128 == 0, N % 128 == 0, K % 32 == 0 (holds for all call sites).
// ---------------------------------------------------------------------------
__global__ __launch_bounds__(256)
void gemm_lrelu_wmma(const float* __restrict__ A,
                     const float* __restrict__ Bm,
                     const float* __restrict__ bias,
                     float* __restrict__ C,
                     int M, int N, int K)
{
    constexpr int TM = 128, TN = 128, TK = 32;
    __shared__ float As[2][TM][TK + 4];   // 2 x 18,432 B
    __shared__ float Bs[2][TK][TN + 4];   // 2 x 16,896 B   (total ~70.7 KB)

    const int tid  = threadIdx.x;
    const int lane = tid & 31;
    const int wave = tid >> 5;
    const int wm   = wave & 3;   // wave row -> rows wm*32 .. wm*32+31
    const int wn   = wave >> 2;  // wave col -> cols wn*64 .. wn*64+63
    const int bm   = blockIdx.y * TM;
    const int bn   = blockIdx.x * TN;

    // WMMA fragment lane mapping (ISA 7.12.2):
    //   A 16x4 : lanes 0-15 -> M=lane,    {K0,K1}; lanes 16-31 -> M=lane-16, {K2,K3}
    //   B 4x16 : lanes 0-15 -> N=lane,    {K0,K1}; lanes 16-31 -> N=lane-16, {K2,K3}
    //   C 16x16: VGPR r, lanes 0-15 -> M=r, N=lane; lanes 16-31 -> M=8+r, N=lane-16
    const int fi = lane & 15;          // M (for A) / N (for B,C)
    const int fk = (lane >> 4) << 1;   // 0 or 2

    // per-thread staging coordinates (4 B128 transfers for A, 4 for B)
    const int a_col  = (tid & 7) << 2;    // 0..28
    const int a_row0 = tid >> 3;          // 0..31  (+p*32)
    const int b_col  = (tid & 31) << 2;   // 0..124
    const int b_row0 = tid >> 5;          // 0..7   (+p*8)

    v8f acc[2][4];
#pragma unroll
    for (int i = 0; i < 2; ++i)
#pragma unroll
        for (int j = 0; j < 4; ++j)
            acc[i][j] = v8f{};

    // stage tile at k0 into LDS buffer `buf` (8 async B128 ops per wave-lane)
    auto stage = [&](int k0, int buf) {
#pragma unroll
        for (int p = 0; p < 4; ++p) {
            const int row = a_row0 + p * 32;
            copy16_to_lds(&A[(size_t)(bm + row) * K + (k0 + a_col)],
                          &As[buf][row][a_col]);
        }
#pragma unroll
        for (int p = 0; p < 4; ++p) {
            const int row = b_row0 + p * 8;
            copy16_to_lds(&Bm[(size_t)(k0 + row) * N + (bn + b_col)],
                          &Bs[buf][row][b_col]);
        }
    };

    stage(0, 0);                       // prologue: tile 0 -> buffer 0
    int cur = 0;

    for (int k0 = 0; k0 < K; k0 += TK) {
        const bool has_next = (k0 + TK) < K;
        if (has_next) {
            stage(k0 + TK, cur ^ 1);   // overlap: tile i+1 in flight
            wait_async<8>();           // tile i's 8 loads retired
        } else {
            wait_async<0>();
        }
        __syncthreads();

#pragma unroll
        for (int kk = 0; kk < TK; kk += 4) {
            v2f a[2], b[4];
#pragma unroll
            for (int i = 0; i < 2; ++i) {
                const int r = wm * 32 + i * 16 + fi;
                a[i].x = As[cur][r][kk + fk];
                a[i].y = As[cur][r][kk + fk + 1];
            }
#pragma unroll
            for (int j = 0; j < 4; ++j) {
                const int c = wn * 64 + j * 16 + fi;
                b[j].x = Bs[cur][kk + fk][c];
                b[j].y = Bs[cur][kk + fk + 1][c];
            }
#pragma unroll
            for (int i = 0; i < 2; ++i)
#pragma unroll
                for (int j = 0; j < 4; ++j)
                    acc[i][j] = __builtin_amdgcn_wmma_f32_16x16x4_f32(
                        /*neg_a=*/false, a[i],
                        /*neg_b=*/false, b[j],
                        /*c_mod=*/(short)0, acc[i][j],
                        /*reuse_a=*/false, /*reuse_b=*/false);
        }
        __syncthreads();               // all waves done with buffer `cur`
        cur ^= 1;
    }

    // ---- epilogue: bias + lrelu + store ----
    const int rbase = (lane >> 4) << 3;   // 0 or 8
#pragma unroll
    for (int i = 0; i < 2; ++i) {
#pragma unroll
        for (int j = 0; j < 4; ++j) {
            const int col = bn + wn * 64 + j * 16 + fi;
            const float bv = bias[col];
#pragma unroll
            for (int r = 0; r < 8; ++r) {
                const int row = bm + wm * 32 + i * 16 + rbase + r;
                C[(size_t)row * N + col] = lrelu(acc[i][j][r] + bv);
            }
        }
    }
}

// ---------------------------------------------------------------------------
// Head GEMM: coeff[b, 0..7] = lrelu(h2[b,:] @ W3 + b3). One wave per row.
// Memory-bound on streaming h2 (256 MB); W3 (128 KB) stays hot in L2.
// ---------------------------------------------------------------------------
__global__ __launch_bounds__(256)
void head_coeff_kernel(const float* __restrict__ h,
                       const float* __restrict__ W3,
                       const float* __restrict__ b3,
                       float* __restrict__ coeff)
{
    const int b    = blockIdx.x * 8 + (threadIdx.x >> 5);
    const int lane = threadIdx.x & 31;
    const float* hrow = h + (size_t)b * HH;

    float acc[8];
#pragma unroll
    for (int j = 0; j < 8; ++j) acc[j] = 0.0f;

    for (int k = lane * 4; k < HH; k += 32 * 4) {
        const float4 hv = *(const float4*)&hrow[k];
        const float hx[4] = {hv.x, hv.y, hv.z, hv.w};
#pragma unroll
        for (int c = 0; c < 4; ++c) {
            const float* wrow = W3 + (size_t)(k + c) * 8;
#pragma unroll
            for (int j = 0; j < 8; ++j) acc[j] += hx[c] * wrow[j];
        }
    }
    // butterfly reduce across the wave32
#pragma unroll
    for (int off = 16; off > 0; off >>= 1)
#pragma unroll
        for (int j = 0; j < 8; ++j)
            acc[j] += __shfl_xor(acc[j], off, 32);

    if (lane == 0) {
        float* crow = coeff + (size_t)b * 8;
#pragma unroll
        for (int j = 0; j < 8; ++j)
            crow[j] = lrelu(acc[j] + b3[j]);
    }
}

// ---------------------------------------------------------------------------
// basis / generate_data / energy. One wave per batch row.
//   basis[b,t,:]  = {1, t, t^2, t^3}
//   gen[b,t,c]    = sum_k basis[b,t,k] * coeff[b, c*4+k]
//   norms[b,k]    = sqrt(sum_t basis[b,t,k]^2)
//   energy[b]     = sum_{c,k} coeff[b,c*4+k]^2 * norms[b,k]
// ---------------------------------------------------------------------------
__global__ __launch_bounds__(256)
void basis_energy_kernel(const float* __restrict__ data_t,
                         const float* __restrict__ coeff,
                         float* __restrict__ gen,
                         float* __restrict__ energy,
                         float* __restrict__ basis)
{
    const int b    = blockIdx.x * 8 + (threadIdx.x >> 5);
    const int lane = threadIdx.x & 31;

    const float* cb = coeff + (size_t)b * 8;
    float c0[4], c1[4];
#pragma unroll
    for (int k = 0; k < 4; ++k) { c0[k] = cb[k]; c1[k] = cb[4 + k]; }

    float s0 = 0.f, s1 = 0.f, s2 = 0.f, s3 = 0.f;  // sum basis[:,k]^2
    for (int t = lane; t < TT; t += 32) {
        const float tv = data_t[(size_t)b * TT + t];
        const float t2 = tv * tv;
        const float t3 = t2 * tv;
        float* bp = basis + ((size_t)b * TT + t) * 4;
        bp[0] = 1.0f; bp[1] = tv; bp[2] = t2; bp[3] = t3;
        float* gp = gen + ((size_t)b * TT + t) * 2;
        gp[0] = c0[0] + c0[1] * tv + c0[2] * t2 + c0[3] * t3;
        gp[1] = c1[0] + c1[1] * tv + c1[2] * t2 + c1[3] * t3;
        s0 += 1.0f; s1 += t2; s2 += t2 * t2; s3 += t3 * t3;
    }
#pragma unroll
    for (int off = 16; off > 0; off >>= 1) {
        s0 += __shfl_xor(s0, off, 32);
        s1 += __shfl_xor(s1, off, 32);
        s2 += __shfl_xor(s2, off, 32);
        s3 += __shfl_xor(s3, off, 32);
    }
    if (lane == 0) {
        const float n[4] = {sqrtf(s0), sqrtf(s1), sqrtf(s2), sqrtf(s3)};
        float e = 0.0f;
#pragma unroll
        for (int k = 0; k < 4; ++k)
            e += (c0[k] * c0[k] + c1[k] * c1[k]) * n[k];
        energy[b] = e;
    }
}

// ---------------------------------------------------------------------------
// Launcher
// ---------------------------------------------------------------------------
extern "C" void kernel_launch(void* const* d_in, const int* in_sizes, int n_in,
                              void* d_out, int out_size, void* d_ws, size_t ws_size,
                              hipStream_t stream)
{
    const float* x      = (const float*)d_in[0];   // (B, 128)
    const float* data_t = (const float*)d_in[1];   // (B, T, 1)
    const float* W1     = (const float*)d_in[2];   // (128, 4096)
    const float* b1     = (const float*)d_in[3];   // (4096,)
    const float* W2     = (const float*)d_in[4];   // (4096, 4096)
    const float* b2     = (const float*)d_in[5];   // (4096,)
    const float* W3     = (const float*)d_in[6];   // (4096, 8)
    const float* b3     = (const float*)d_in[7];   // (8,)

    float* out    = (float*)d_out;
    float* gen    = out;                                   // B*T*2
    float* energy = gen + (size_t)BB * TT * 2;             // B
    float* coeff  = energy + (size_t)BB;                   // B*8
    float* basis  = coeff + (size_t)BB * 8;                // B*T*4

    float* h1 = (float*)d_ws;                              // B*H floats (256 MB)
    float* h2 = h1 + (size_t)BB * HH;                      // B*H floats (256 MB)

    const dim3 blk(256);

    // Layer 1: h1 = lrelu(x @ W1 + b1)        [M=B, N=H, K=ZDIM]
    gemm_lrelu_wmma<<<dim3(HH / 128, BB / 128), blk, 0, stream>>>(
        x, W1, b1, h1, BB, HH, ZDIM);

    // Layer 2: h2 = lrelu(h1 @ W2 + b2)       [M=B, N=H, K=H]  (dominant)
    gemm_lrelu_wmma<<<dim3(HH / 128, BB / 128), blk, 0, stream>>>(
        h1, W2, b2, h2, BB, HH, HH);

    // Layer 3: coeff = lrelu(h2 @ W3 + b3)    (written straight into d_out)
    head_coeff_kernel<<<BB / 8, blk, 0, stream>>>(h2, W3, b3, coeff);

    // basis / generate_data / energy
    basis_energy_kernel<<<BB / 8, blk, 0, stream>>>(data_t, coeff, gen, energy, basis);
}